// GCNLayer_55671366090796
// MI455X (gfx1250) — compile-verified
//
#include <hip/hip_runtime.h>
#include <hip/hip_bf16.h>

typedef __attribute__((ext_vector_type(2))) float v2f;
typedef __attribute__((ext_vector_type(8))) float v8f;

#define IN_FEATS  256
#define OUT_FEATS 128

// ---------------------------------------------------------------------------
// 1) degree histogram over row endpoints (self loop added analytically later)
// ---------------------------------------------------------------------------
__global__ void gcn_degree_kernel(const int* __restrict__ ei, float* __restrict__ deg, int E) {
    int e = blockIdx.x * blockDim.x + threadIdx.x;
    if (e < E) {
        atomicAdd(&deg[ei[e]], 1.0f);   // rows are ei[0..E)
    }
}

// ---------------------------------------------------------------------------
// 2) dis[i] = rsqrt(deg[i] + 1)   (+1 = self loop; deg >= 1 always)
// ---------------------------------------------------------------------------
__global__ void gcn_rsqrt_kernel(float* __restrict__ deg, int N) {
    int i = blockIdx.x * blockDim.x + threadIdx.x;
    if (i < N) {
        deg[i] = rsqrtf(deg[i] + 1.0f);
    }
}

// ---------------------------------------------------------------------------
// 3) h = x @ W + b  via V_WMMA_F32_16X16X4_F32
//    block = 256 threads = 8 waves; block owns rows [16*bid, 16*bid+16)
//    wave w owns output cols [16w, 16w+16)  (OUT_FEATS = 128 = 8 tiles)
//    A 16x4 layout: lanes 0-15 -> {K, K+1}, lanes 16-31 -> {K+2, K+3}, M = lane%16
//    B 4x16 layout: lanes 0-15 -> rows {K, K+1}, lanes 16-31 -> rows {K+2, K+3}, N = lane%16
//    C/D 16x16:     VGPR r: lanes 0-15 M=r, lanes 16-31 M=r+8, N = lane%16
//    Bias trick: every C element in VGPR r / lane L has column N = L%16, so the
//    accumulator is simply *initialized* with bias[col0 + L%16] in all 8 VGPRs.
// ---------------------------------------------------------------------------
__global__ __launch_bounds__(256) void gcn_gemm_wmma_kernel(
        const float* __restrict__ x, const float* __restrict__ W,
        const float* __restrict__ bias, float* __restrict__ h, int N) {
    const int wave  = threadIdx.x >> 5;          // 0..7
    const int lane  = threadIdx.x & 31;
    const int l16   = lane & 15;
    const int lhalf = lane >> 4;                 // 0 or 1
    const int row0  = blockIdx.x * 16;
    const int col0  = wave * 16;
    if (row0 >= N) return;

    int rowA = row0 + l16;
    if (rowA >= N) rowA = N - 1;                 // clamp: keep EXEC all-ones for WMMA

    // A source: x[rowA][k + 2*lhalf], x[rowA][k + 2*lhalf + 1]
    const float* xrow = x + (size_t)rowA * IN_FEATS + 2 * lhalf;
    // B source: W[k + 2*lhalf][col0 + l16], W[k + 2*lhalf + 1][col0 + l16]
    const float* wcol = W + (size_t)(2 * lhalf) * OUT_FEATS + (col0 + l16);

    // accumulator pre-loaded with bias (column depends only on lane)
    const float bv = bias[col0 + l16];
    v8f acc = {bv, bv, bv, bv, bv, bv, bv, bv};

    #pragma unroll 8
    for (int k = 0; k < IN_FEATS; k += 4) {
        v2f a;
        a.x = xrow[k];
        a.y = xrow[k + 1];
        v2f b;
        b.x = wcol[(size_t)k * OUT_FEATS];
        b.y = wcol[(size_t)k * OUT_FEATS + OUT_FEATS];
        // 8 args: (neg_a, A, neg_b, B, c_mod, C, reuse_a, reuse_b)
        acc = __builtin_amdgcn_wmma_f32_16x16x4_f32(
                false, a, false, b, (short)0, acc, false, false);
    }

    float* hp = h + ((size_t)row0 + 8 * lhalf) * OUT_FEATS + col0 + l16;
    if (row0 + 16 <= N) {
        // full tile (always the case for N % 16 == 0): branch-free stores
        #pragma unroll
        for (int r = 0; r < 8; ++r) {
            hp[(size_t)r * OUT_FEATS] = acc[r];
        }
    } else {
        #pragma unroll
        for (int r = 0; r < 8; ++r) {
            if (row0 + r + 8 * lhalf < N) {
                hp[(size_t)r * OUT_FEATS] = acc[r];
            }
        }
    }
}

// ---------------------------------------------------------------------------
// 4) edge scatter: one wave per edge; 32 lanes x float4 = 128 features
//    out[row] += dis[row]*dis[col] * h[col]   (atomic f32, lands in L2)
// ---------------------------------------------------------------------------
__global__ __launch_bounds__(256) void gcn_scatter_kernel(
        const int* __restrict__ ei, const float* __restrict__ dis,
        const float* __restrict__ h, float* __restrict__ out, int E) {
    const int e = blockIdx.x * (blockDim.x >> 5) + (threadIdx.x >> 5);
    if (e >= E) return;
    const int lane = threadIdx.x & 31;
    const int r = ei[e];
    const int c = ei[E + e];
    const float w = dis[r] * dis[c];

    const float4 v = ((const float4*)(h + (size_t)c * OUT_FEATS))[lane];
    float* op = out + (size_t)r * OUT_FEATS + lane * 4;
    atomicAdd(op + 0, w * v.x);
    atomicAdd(op + 1, w * v.y);
    atomicAdd(op + 2, w * v.z);
    atomicAdd(op + 3, w * v.w);
}

// ---------------------------------------------------------------------------
// 5) self loops: out[i] += dis[i]^2 * h[i]  (plain RMW; runs after scatter)
// ---------------------------------------------------------------------------
__global__ __launch_bounds__(256) void gcn_selfloop_kernel(
        const float* __restrict__ dis, const float* __restrict__ h,
        float* __restrict__ out, int N) {
    const int i = blockIdx.x * (blockDim.x >> 5) + (threadIdx.x >> 5);
    if (i >= N) return;
    const int lane = threadIdx.x & 31;
    const float w = dis[i] * dis[i];

    const float4 v = ((const float4*)(h + (size_t)i * OUT_FEATS))[lane];
    float4* op = (float4*)(out + (size_t)i * OUT_FEATS);
    float4 o = op[lane];
    o.x += w * v.x;
    o.y += w * v.y;
    o.z += w * v.z;
    o.w += w * v.w;
    op[lane] = o;
}

// ---------------------------------------------------------------------------
extern "C" void kernel_launch(void* const* d_in, const int* in_sizes, int n_in,
                              void* d_out, int out_size, void* d_ws, size_t ws_size,
                              hipStream_t stream) {
    const float* x    = (const float*)d_in[0];   // [N, 256]
    const int*   ei   = (const int*)d_in[1];     // [2, E] flat
    const float* W    = (const float*)d_in[2];   // [256, 128]
    const float* bias = (const float*)d_in[3];   // [128]
    float*       out  = (float*)d_out;           // [N, 128]

    const int N = in_sizes[0] / IN_FEATS;        // 100000
    const int E = in_sizes[1] / 2;               // 3200000

    // workspace layout: h [N*128 f32] | deg/dis [N f32]
    float* h   = (float*)d_ws;
    float* deg = h + (size_t)N * OUT_FEATS;

    hipMemsetAsync(out, 0, (size_t)N * OUT_FEATS * sizeof(float), stream);
    hipMemsetAsync(deg, 0, (size_t)N * sizeof(float), stream);

    gcn_degree_kernel<<<(E + 255) / 256, 256, 0, stream>>>(ei, deg, E);
    gcn_rsqrt_kernel<<<(N + 255) / 256, 256, 0, stream>>>(deg, N);
    gcn_gemm_wmma_kernel<<<(N + 15) / 16, 256, 0, stream>>>(x, W, bias, h, N);
    gcn_scatter_kernel<<<(E + 7) / 8, 256, 0, stream>>>(ei, deg, h, out, E);
    gcn_selfloop_kernel<<<(N + 7) / 8, 256, 0, stream>>>(deg, h, out, N);
}